// SISDLFEmbeddingLoss_919123001994
// MI455X (gfx1250) — compile-verified
//
#include <hip/hip_runtime.h>

#define B_IMGS 8
#define CH 32
#define HWPIX (512*512)
#define DDv 2.5f

typedef __attribute__((ext_vector_type(2))) float v2f;
typedef __attribute__((ext_vector_type(8))) float v8f;
typedef __attribute__((ext_vector_type(4))) unsigned v4u;
typedef __attribute__((ext_vector_type(8))) int v8i;
typedef __attribute__((ext_vector_type(4))) int v4i;

// workspace layout (in dwords)
#define WS_SUMS  0      // 8*8*32 floats
#define WS_MEANS 2048   // 8*8*32 floats
#define WS_DSUM  4096   // 8*8 floats
#define WS_DLOSS 4160   // 8 floats
#define WS_CNT   4168   // 8*8 ints
#define WS_TOTAL 4232

#define TROW 132          // padded LDS row stride (floats): all-64-bank B reads
#define TPIX 128          // pixels per tile
#define NTILES 64         // tiles per block (8192 pixels)

#if __has_builtin(__builtin_amdgcn_tensor_load_to_lds)
#define USE_TDM 1
#else
#define USE_TDM 0
#endif

__global__ void k0_init(float* ws) {
    int i = blockIdx.x * blockDim.x + threadIdx.x;
    if (i < WS_TOTAL) ws[i] = 0.0f;   // 0.0f bit pattern also zeroes the int counts
}

#if !USE_TDM
__device__ __forceinline__ void async_b128(unsigned lds_off, unsigned goff, const float* base) {
    asm volatile("global_load_async_to_lds_b128 %0, %1, %2"
                 :: "v"(lds_off), "v"(goff), "s"(base)
                 : "memory");
}
__device__ __forceinline__ void wait_async0() {
    asm volatile("s_wait_asynccnt 0" ::: "memory");
}
#endif

// Pass 1: per-class channel sums via V_WMMA_F32_16X16X4_F32 (one-hot x embeddings).
// Tile staging: TENSOR_LOAD_TO_LDS (TDM, one DMA per 16KB tile, row-padded by pad_enable)
// when available, else double-buffered GLOBAL_LOAD_ASYNC_TO_LDS_B128.
// Grid: (32 chunks, 8 images), 256 threads (8 waves).
__global__ __launch_bounds__(256) void k1_sums(const float* __restrict__ emb,
                                               const int* __restrict__ lab,
                                               float* __restrict__ sums_g,
                                               int* __restrict__ cnt_g) {
    __shared__ float tile_s[2][CH * TROW];   // [buf][c*132 + pix]  (DMA target)
    __shared__ int   lab_s[2][TPIX];
    __shared__ int   cnt_s[8];
    __shared__ float red_s[8 * 256];

    const int t    = threadIdx.x;
    const int b    = blockIdx.y;
    const int chnk = blockIdx.x;         // 0..31, 8192 pixels each
    const int wave = t >> 5;
    const int lane = t & 31;
    const int n    = lane & 15;          // class row (A) / channel col (B)
    const int hf   = lane >> 4;          // 0: K=0,1 ; 1: K=2,3

    const float* embB = emb + (size_t)b * CH * HWPIX;
    const int*   labB = lab + (size_t)b * HWPIX;

    if (t < 8) cnt_s[t] = 0;
    __syncthreads();

    v8f acc0 = {};   // classes x channels 0..15
    v8f acc1 = {};   // classes x channels 16..31

    // ---- issue helper (tile -> buf): labels sync, embeddings DMA ----
    auto issue = [&](int tl, int buf) {
        const int p0 = chnk * (NTILES * TPIX) + tl * TPIX;
        if (t < TPIX) {
            int L = labB[p0 + t];
            lab_s[buf][t] = L;
            atomicAdd(&cnt_s[L], 1);
        }
#if USE_TDM
        if (wave == 0) {   // TDM ignores EXEC: exactly one wave issues the descriptor
            const unsigned lds_addr = (unsigned)(uintptr_t)&tile_s[buf][0];
            const unsigned long long ga = (unsigned long long)(uintptr_t)(embB + p0);
            v4u g0;
            g0[0] = 1u;                                   // count=1, user mode
            g0[1] = lds_addr;                             // LDS byte address
            g0[2] = (unsigned)ga;                         // global_addr[31:0]
            g0[3] = (unsigned)(ga >> 32) | (2u << 30);    // global_addr[56:32] | type=2
            v8i g1;
            g1[0] = (int)((2u << 16)      // data_size = 4B
                        | (1u << 20)      // pad_enable
                        | (6u << 22)      // pad_interval: every 128 dwords
                        | (3u << 25));    // pad_amount: 4 dwords -> row stride 132
            g1[1] = (int)(128u << 16);    // tensor_dim0 = 128 (low 16)
            g1[2] = (int)(32u << 16);     // tensor_dim0 hi=0 | tensor_dim1 = 32 (low 16)
            g1[3] = (int)(128u << 16);    // tensor_dim1 hi=0 | tile_dim0 = 128
            g1[4] = 32;                   // tile_dim1 = 32, tile_dim2 = 0
            g1[5] = HWPIX;                // tensor_dim0_stride[31:0]
            g1[6] = 0;                    // stride hi | tensor_dim1_stride lo
            g1[7] = 0;
            v4i zz = {0, 0, 0, 0};        // 2D tensor: groups 2/3 unused (zeros)
#if __clang_major__ >= 23
            v8i z8 = {0, 0, 0, 0, 0, 0, 0, 0};
            __builtin_amdgcn_tensor_load_to_lds(g0, g1, zz, zz, z8, 0);
#else
            __builtin_amdgcn_tensor_load_to_lds(g0, g1, zz, zz, 0);
#endif
        }
#else
        const unsigned ldsbase = (unsigned)(uintptr_t)&tile_s[buf][0];
        #pragma unroll
        for (int i = 0; i < 4; ++i) {                      // 4 x 256thr x 16B = 16KB tile
            const int c    = 8 * i + (t >> 5);             // channel
            const int pix4 = lane * 4;                     // pixel quad
            const unsigned goff = (unsigned)((c * HWPIX + p0 + pix4) * 4);
            const unsigned loff = ldsbase + (unsigned)((c * TROW + pix4) * 4);
            async_b128(loff, goff, embB);
        }
#endif
    };

    issue(0, 0);
    int buf = 0;
    for (int tl = 0; tl < NTILES; ++tl) {
#if USE_TDM
        if (wave == 0) __builtin_amdgcn_s_wait_tensorcnt(0);
#else
        wait_async0();                   // my async stores to LDS landed
#endif
        __syncthreads();                 // tile[buf] visible to all waves
        if (tl + 1 < NTILES) issue(tl + 1, buf ^ 1);   // overlap next tile's DMA

        // ---- each wave consumes its 16 pixels in K-chunks of 4 ----
        const int pw = wave * 16;
        const float* tb = &tile_s[buf][0];
        #pragma unroll
        for (int kc = 0; kc < 4; ++kc) {
            const int k0 = pw + kc * 4 + 2 * hf;  // this lane's K rows: k0, k0+1
            v2f a, b0, b1;
            a.x = (lab_s[buf][k0]     == n) ? 1.0f : 0.0f;   // A[m=n][K]
            a.y = (lab_s[buf][k0 + 1] == n) ? 1.0f : 0.0f;
            b0.x = tb[n * TROW + k0];                        // B[K][N=n], ch 0..15
            b0.y = tb[n * TROW + k0 + 1];
            b1.x = tb[(16 + n) * TROW + k0];                 // ch 16..31
            b1.y = tb[(16 + n) * TROW + k0 + 1];
            acc0 = __builtin_amdgcn_wmma_f32_16x16x4_f32(false, a, false, b0,
                                                         (short)0, acc0, false, false);
            acc1 = __builtin_amdgcn_wmma_f32_16x16x4_f32(false, a, false, b1,
                                                         (short)0, acc1, false, false);
        }
        buf ^= 1;
    }
    __syncthreads();
    // D layout: class r -> acc VGPR r, lanes 0..15 = channel columns
    if (lane < 16) {
        #pragma unroll
        for (int r = 0; r < 8; ++r) {
            red_s[wave * 256 + r * 32 + n]      = acc0[r];
            red_s[wave * 256 + r * 32 + 16 + n] = acc1[r];
        }
    }
    __syncthreads();
    float tot = 0.0f;
    #pragma unroll
    for (int w = 0; w < 8; ++w) tot += red_s[w * 256 + t];
    atomicAdd(&sums_g[b * 256 + t], tot);
    if (t < 8) atomicAdd(&cnt_g[b * 8 + t], cnt_s[t]);
}

// means + pairwise hinge loss between means of classes 1..7. Grid: 8 blocks x 256.
__global__ __launch_bounds__(256) void k2_means(const float* __restrict__ sums_g,
                                                const int* __restrict__ cnt_g,
                                                float* __restrict__ means_g,
                                                float* __restrict__ dloss_g) {
    __shared__ float m_s[256];
    __shared__ float dl_s;
    const int t = threadIdx.x;
    const int b = blockIdx.x;
    if (t == 0) dl_s = 0.0f;
    float mv = sums_g[b * 256 + t] / (float)cnt_g[b * 8 + (t >> 5)];
    means_g[b * 256 + t] = mv;
    m_s[t] = mv;
    __syncthreads();
    if (t < 21) {
        int pi = 1, pj = 2, idx = 0;
        for (int i = 1; i < 8; ++i)
            for (int j = i + 1; j < 8; ++j) {
                if (idx == t) { pi = i; pj = j; }
                ++idx;
            }
        float sq = 0.0f;
        for (int c = 0; c < 32; ++c) {
            float d = m_s[pi * 32 + c] - m_s[pj * 32 + c];
            sq = fmaf(d, d, sq);
        }
        float dist = sqrtf(sq);
        float marg = 2.0f * DDv - dist;
        float h = (dist < 2.0f * DDv) ? marg * marg : 0.0f;
        atomicAdd(&dl_s, h);
    }
    __syncthreads();
    if (t == 0) dloss_g[b] = dl_s / 6.0f;
}

// Pass 2: per-pixel ||e - mean[label]||, segment-summed by class.
// Image order reversed vs pass 1 so the L2-resident (most recently streamed)
// images are re-read first: most of this pass hits the 192MB L2, not HBM.
// Grid: (16 chunks, 8 images), 256 threads, 4 pixels/thread/iter (float4 loads).
__global__ __launch_bounds__(256) void k3_dist(const float* __restrict__ emb,
                                               const int* __restrict__ lab,
                                               const float* __restrict__ means_g,
                                               float* __restrict__ dsum_g) {
    __shared__ float mean_s[8 * 33];   // stride 33 -> per-label broadcast, no conflicts
    __shared__ float dsum_s[8];
    const int t    = threadIdx.x;
    const int b    = (B_IMGS - 1) - blockIdx.y;   // reverse order for L2 reuse
    const int chnk = blockIdx.x;       // 0..15, 16384 pixels each
    if (t < 8) dsum_s[t] = 0.0f;
    {
        int k = t >> 5, c = t & 31;
        mean_s[k * 33 + c] = means_g[b * 256 + t];
    }
    __syncthreads();
    const float* embB = emb + (size_t)b * CH * HWPIX;
    const int*   labB = lab + (size_t)b * HWPIX;
    for (int i = 0; i < 16; ++i) {
        const int p = chnk * 16384 + i * 1024 + t * 4;
        int4 L4 = *reinterpret_cast<const int4*>(&labB[p]);
        const float* m0 = &mean_s[L4.x * 33];
        const float* m1 = &mean_s[L4.y * 33];
        const float* m2 = &mean_s[L4.z * 33];
        const float* m3 = &mean_s[L4.w * 33];
        float s0 = 0.f, s1 = 0.f, s2 = 0.f, s3 = 0.f;
        #pragma unroll
        for (int c = 0; c < CH; ++c) {
            float4 v = *reinterpret_cast<const float4*>(&embB[(size_t)c * HWPIX + p]);
            float d0 = v.x - m0[c]; s0 = fmaf(d0, d0, s0);
            float d1 = v.y - m1[c]; s1 = fmaf(d1, d1, s1);
            float d2 = v.z - m2[c]; s2 = fmaf(d2, d2, s2);
            float d3 = v.w - m3[c]; s3 = fmaf(d3, d3, s3);
        }
        atomicAdd(&dsum_s[L4.x], sqrtf(s0));
        atomicAdd(&dsum_s[L4.y], sqrtf(s1));
        atomicAdd(&dsum_s[L4.z], sqrtf(s2));
        atomicAdd(&dsum_s[L4.w], sqrtf(s3));
    }
    __syncthreads();
    if (t < 8) atomicAdd(&dsum_g[b * 8 + t], dsum_s[t]);
}

__global__ void k4_final(const float* __restrict__ dsum_g,
                         const int* __restrict__ cnt_g,
                         const float* __restrict__ dloss_g,
                         float* __restrict__ out) {
    if (threadIdx.x == 0 && blockIdx.x == 0) {
        float acc = 0.0f;
        for (int b = 0; b < 8; ++b) {
            float vl = 0.0f;
            for (int k = 1; k < 8; ++k)
                vl += dsum_g[b * 8 + k] / (float)cnt_g[b * 8 + k];
            acc += (vl + dloss_g[b]) / 7.0f;
        }
        out[0] = acc / 8.0f;
    }
}

extern "C" void kernel_launch(void* const* d_in, const int* in_sizes, int n_in,
                              void* d_out, int out_size, void* d_ws, size_t ws_size,
                              hipStream_t stream) {
    (void)in_sizes; (void)n_in; (void)out_size; (void)ws_size;
    const float* emb = (const float*)d_in[0];
    const int*   lab = (const int*)d_in[1];
    float* out = (float*)d_out;
    float* ws  = (float*)d_ws;

    float* sums_g  = ws + WS_SUMS;
    float* means_g = ws + WS_MEANS;
    float* dsum_g  = ws + WS_DSUM;
    float* dloss_g = ws + WS_DLOSS;
    int*   cnt_g   = (int*)(ws + WS_CNT);

    k0_init<<<(WS_TOTAL + 255) / 256, 256, 0, stream>>>(ws);
    k1_sums<<<dim3(32, 8), 256, 0, stream>>>(emb, lab, sums_g, cnt_g);
    k2_means<<<8, 256, 0, stream>>>(sums_g, cnt_g, means_g, dloss_g);
    k3_dist<<<dim3(16, 8), 256, 0, stream>>>(emb, lab, means_g, dsum_g);
    k4_final<<<1, 64, 0, stream>>>(dsum_g, cnt_g, dloss_g, out);
}